// OrthogonalLayer1D_56203942036017
// MI455X (gfx1250) — compile-verified
//
#include <hip/hip_runtime.h>
#include <math.h>

typedef float v2f __attribute__((ext_vector_type(2)));
typedef float v8f __attribute__((ext_vector_type(8)));

#define DIM  512
#define NEXP 8

// One wave32 per block; each block orthonormalizes TWO samples (16 stacked rows).
// Classical Gram-Schmidt == Q = L^{-1} X with G = X X^T = L L^T (Cholesky).
__global__ __launch_bounds__(32)
void gs_cholesky_wmma(const float* __restrict__ x, float* __restrict__ out,
                      unsigned nSamples) {
  const int lane    = threadIdx.x;       // 0..31
  const unsigned s0 = blockIdx.x * 2u;   // first of the two samples

  __shared__ float    Gm[16 * 16];
  __shared__ float    Wm[16 * 16];
  __shared__ unsigned rowOff[16];        // element offset of each stacked row in x

  // x layout: [E=8, T, B, D] ; sample s = t*B+b ; row(e,s) at (e*nSamples + s)*DIM
  if (lane < 16) {
    const unsigned e = (unsigned)lane & 7u;
    const unsigned s = s0 + (unsigned)(lane >> 3);
    rowOff[lane] = (e * nSamples + s) * (unsigned)DIM;
  }
#pragma unroll
  for (int i = 0; i < 8; ++i) Wm[lane + 32 * i] = 0.0f;  // block-diagonal W: zero rest
  __syncthreads();

  const int half = lane >> 4;   // 0 | 1 (lane half)
  const int mrow = lane & 15;   // stacked row id for A fragments, N id for B and C/D
  const int koff = half << 1;   // K sub-offset 0 | 2 (f32 16x16x4 layout)

  const float* prow = x + (size_t)rowOff[mrow];

  // ---------- Phase A: G = Xp * Xp^T  (Xp = stacked 16 x 512) ----------
  // For 16x16x4 f32, the A-frag and B-frag of X / X^T are identical per lane:
  // lane holds {X[mrow][4t+koff], X[mrow][4t+koff+1]}.
  v8f g = {0.f, 0.f, 0.f, 0.f, 0.f, 0.f, 0.f, 0.f};
#pragma unroll 4
  for (int t = 0; t < DIM / 4; ++t) {
    const v2f a = *(const v2f*)(prow + 4 * t + koff);   // 8B, 8B-aligned
    g = __builtin_amdgcn_wmma_f32_16x16x4_f32(false, a, false, a,
                                              (short)0, g, false, false);
  }
  // C/D layout: VGPR r -> row (r + 8*half), col mrow
#pragma unroll
  for (int r = 0; r < 8; ++r) Gm[(r + half * 8) * 16 + mrow] = g[r];
  __syncthreads();

  // ---------- Phase B: per-sample 8x8 Cholesky (in place) + W = L^{-1} ----------
  if (lane < 2) {
    const int o = lane * 8;   // diagonal block of this lane's sample
    for (int i = 0; i < 8; ++i) {
      float d = Gm[(o + i) * 16 + (o + i)];
      for (int k = 0; k < i; ++k) {
        const float v = Gm[(o + i) * 16 + (o + k)];
        d -= v * v;
      }
      d = sqrtf(d);
      Gm[(o + i) * 16 + (o + i)] = d;
      const float inv = 1.0f / d;
      for (int j = i + 1; j < 8; ++j) {
        float acc = Gm[(o + j) * 16 + (o + i)];
        for (int k = 0; k < i; ++k)
          acc -= Gm[(o + j) * 16 + (o + k)] * Gm[(o + i) * 16 + (o + k)];
        Gm[(o + j) * 16 + (o + i)] = acc * inv;
      }
    }
    // forward-substitute columns of I: W = L^{-1} (lower triangular)
    for (int j = 0; j < 8; ++j) {
      Wm[(o + j) * 16 + (o + j)] = 1.0f / Gm[(o + j) * 16 + (o + j)];
      for (int i = j + 1; i < 8; ++i) {
        float acc = 0.0f;
        for (int k = j; k < i; ++k)
          acc += Gm[(o + i) * 16 + (o + k)] * Wm[(o + k) * 16 + (o + j)];
        Wm[(o + i) * 16 + (o + j)] = -acc / Gm[(o + i) * 16 + (o + i)];
      }
    }
  }
  __syncthreads();

  // ---------- Phase C: Q = W * Xp, streamed over 32 N-tiles of dim ----------
  // A-frag of W (K = 16, 4 K-tiles): lane holds {W[mrow][4tk+koff], W[mrow][4tk+koff+1]}
  v2f aw[4];
  unsigned br0[4], br1[4];
#pragma unroll
  for (int tk = 0; tk < 4; ++tk) {
    const int k0 = 4 * tk + koff;
    aw[tk].x = Wm[mrow * 16 + k0];
    aw[tk].y = Wm[mrow * 16 + k0 + 1];
    br0[tk] = rowOff[k0];      // B-frag rows k0, k0+1 at column (16n + mrow)
    br1[tk] = rowOff[k0 + 1];
  }

  for (int n = 0; n < DIM / 16; ++n) {
    const int col = n * 16 + mrow;
    v8f q = {0.f, 0.f, 0.f, 0.f, 0.f, 0.f, 0.f, 0.f};
#pragma unroll
    for (int tk = 0; tk < 4; ++tk) {
      v2f b;
      b.x = x[(size_t)br0[tk] + col];   // wave-wide: 2 x 64B contiguous segments
      b.y = x[(size_t)br1[tk] + col];
      q = __builtin_amdgcn_wmma_f32_16x16x4_f32(false, aw[tk], false, b,
                                                (short)0, q, false, false);
    }
    // out layout: [T, B, E, D] -> ((s*8 + e)*512 + d). Write-once: non-temporal.
#pragma unroll
    for (int r = 0; r < 8; ++r) {
      const unsigned M = (unsigned)(r + half * 8);
      const unsigned s = s0 + (M >> 3);
      const unsigned e = M & 7u;
      __builtin_nontemporal_store(q[r], out + ((size_t)s * NEXP + e) * DIM + col);
    }
  }
}

extern "C" void kernel_launch(void* const* d_in, const int* in_sizes, int n_in,
                              void* d_out, int out_size, void* d_ws, size_t ws_size,
                              hipStream_t stream) {
  (void)n_in; (void)out_size; (void)d_ws; (void)ws_size;
  const float* x = (const float*)d_in[0];
  float* out = (float*)d_out;

  const unsigned total    = (unsigned)in_sizes[0];          // E*T*B*D
  const unsigned nSamples = total / (NEXP * DIM);           // T*B = 8192
  const unsigned nBlocks  = nSamples / 2;                   // 2 samples per wave

  gs_cholesky_wmma<<<dim3(nBlocks), dim3(32), 0, stream>>>(x, out, nSamples);
}